// AdditiveAttention_17282948399623
// MI455X (gfx1250) — compile-verified
//
#include <hip/hip_runtime.h>
#include <hip/hip_bf16.h>

// Problem constants (match reference)
#define B_  16
#define Q_  512
#define K_  512
#define H_  64
#define D_  256
#define MASK_VAL (-1000000.0f)
#define KPAD 68    // LDS row stride (floats) for 64-wide rows: 272B, 16B-aligned
#define APAD 516   // LDS row stride (floats) for 512-wide attn rows: 2064B, 16B-aligned
#define VPAD 68    // LDS row stride (floats) for 64-wide values rows

typedef __attribute__((ext_vector_type(2))) float v2f;
typedef __attribute__((ext_vector_type(8))) float v8f;
typedef __attribute__((ext_vector_type(4))) int   v4i;

#if __has_builtin(__builtin_amdgcn_global_load_async_to_lds_b128)
#define HAVE_ASYNC_LDS 1
#else
#define HAVE_ASYNC_LDS 0
#endif

#if HAVE_ASYNC_LDS
typedef __attribute__((address_space(1))) v4i* gv4i_p;   // global v4i*
typedef __attribute__((address_space(3))) v4i* lv4i_p;   // LDS v4i*
#endif

// 16-byte global -> LDS copy. Async DMA path (ASYNCcnt) on gfx1250, VGPR
// round-trip fallback otherwise.
__device__ __forceinline__ void stage16(const float* g, float* l) {
#if HAVE_ASYNC_LDS
    float* gnc = const_cast<float*>(g);
    __builtin_amdgcn_global_load_async_to_lds_b128(
        (gv4i_p)gnc, (lv4i_p)l, 0, 0);
#else
    *(float4*)l = *(const float4*)g;
#endif
}

// Complete all outstanding async LDS fills of this wave, then workgroup barrier.
__device__ __forceinline__ void stage_commit() {
#if HAVE_ASYNC_LDS
#if __has_builtin(__builtin_amdgcn_s_wait_asynccnt)
    __builtin_amdgcn_s_wait_asynccnt(0);
#else
    asm volatile("s_wait_asynccnt 0" ::: "memory");
#endif
#endif
    __syncthreads();
}

__device__ __forceinline__ float fast_tanh(float x) {
#if __has_builtin(__builtin_amdgcn_tanhf)
    return __builtin_amdgcn_tanhf(x);          // gfx1250 native v_tanh_f32
#else
    float e = __expf(2.0f * x);                // v_exp_f32 + v_rcp path
    return 1.0f - 2.0f / (e + 1.0f);
#endif
}

// ---------------------------------------------------------------------------
// Kernel 1: q/k projections via V_WMMA_F32_16X16X4_F32.
// One wave computes a 16(rows) x 16(h) tile, K-reduction 256 in steps of 4.
// Blocks [0, 512): queries -> qf ; blocks [512, 1024): keys -> kf.
// ---------------------------------------------------------------------------
__global__ __launch_bounds__(128)
void proj_kernel(const float* __restrict__ queries,
                 const float* __restrict__ keys,
                 const float* __restrict__ Wq,
                 const float* __restrict__ Wk,
                 float* __restrict__ qf,
                 float* __restrict__ kf)
{
    const int NQ_TILES = (B_ * Q_) / 16;       // 512
    int t = blockIdx.x;
    bool isQ = (t < NQ_TILES);
    int rt = isQ ? t : (t - NQ_TILES);
    const float* X = isQ ? queries : keys;
    const float* W = isQ ? Wq : Wk;            // [H, 256] row-major (weight [out,in])
    float* Y = isQ ? qf : kf;                  // [rows, H] row-major

    int lane  = threadIdx.x & 31;
    int wave  = threadIdx.x >> 5;
    int hbase = wave * 16;                     // 4 waves cover H=64
    int mn    = lane & 15;
    int khalf = (lane >> 4) << 1;              // lanes 16-31 hold K=2,3

    size_t rowbase = (size_t)rt * 16;
    const float* Xrow = X + (rowbase + mn) * D_;
    const float* Wn   = W + (size_t)(hbase + mn) * D_;   // B tile: Wt[k][n] = W[n*256+k]

    v8f acc = {};
    #pragma unroll 4
    for (int k0 = 0; k0 < D_; k0 += 4) {
        v2f a = *(const v2f*)(Xrow + k0 + khalf);        // A 16x4 tile
        v2f b = *(const v2f*)(Wn   + k0 + khalf);        // B 4x16 tile
        acc = __builtin_amdgcn_wmma_f32_16x16x4_f32(
                  false, a, false, b, (short)0, acc, false, false);
    }

    #pragma unroll
    for (int r = 0; r < 8; ++r) {
        int m = r + ((lane >> 4) << 3);                  // C/D: M = r (+8 for hi half)
        Y[(rowbase + m) * H_ + hbase + mn] = acc[r];
    }
}

// ---------------------------------------------------------------------------
// Kernel 2: fused additive scores + mask + softmax (transcendental-bound).
// One block = (batch b, 16 q-rows). k-features staged in LDS via async DMA in
// 4 chunks of 128 rows. Each wave owns 2 q-rows; each lane holds 16 scores in
// regs, then wave32 shfl_xor softmax. Scores never touch HBM.
// ---------------------------------------------------------------------------
__global__ __launch_bounds__(256)
void score_softmax_kernel(const float* __restrict__ qf,
                          const float* __restrict__ kf,
                          const float* __restrict__ wv_g,
                          const int*   __restrict__ valid_lens,
                          float* __restrict__ attn)
{
    __shared__ float skf[128 * KPAD];   // k-feature chunk [128][64] padded
    __shared__ float sqf[16 * KPAD];    // q-feature tile  [16][64] padded
    __shared__ float swv[64];

    int b   = blockIdx.x >> 5;          // 32 q-tiles per batch
    int qt  = blockIdx.x & 31;
    int tid  = threadIdx.x;
    int lane = tid & 31;
    int wave = tid >> 5;

    {   // stage q tile (exactly one float4 per thread) + wv (async, completes
        // at the first stage_commit below)
        int r = tid >> 4, j = tid & 15;
        stage16(qf + ((size_t)(b * Q_) + qt * 16 + r) * H_ + j * 4,
                &sqf[r * KPAD + j * 4]);
        if (tid < 16)
            stage16(wv_g + tid * 4, &swv[tid * 4]);
    }

    float sc[2][16];                    // per-lane scores for this wave's 2 rows
    int row0 = wave * 2;

    for (int c = 0; c < 4; ++c) {       // 4 chunks of 128 k-rows
        __syncthreads();                // protect previous chunk's LDS reads
        #pragma unroll
        for (int idx = tid; idx < 128 * 16; idx += 256) {
            int r = idx >> 4, j = idx & 15;
            stage16(kf + ((size_t)(b * K_) + c * 128 + r) * H_ + j * 4,
                    &skf[r * KPAD + j * 4]);
        }
        stage_commit();                 // s_wait_asynccnt 0 + barrier

        for (int rr = 0; rr < 2; ++rr) {
            int row = row0 + rr;
            for (int i = 0; i < 4; ++i) {               // k = c*128 + i*32 + lane
                const float* kr = &skf[(i * 32 + lane) * KPAD];
                const float* qr = &sqf[row * KPAD];
                float s = 0.0f;
                #pragma unroll
                for (int h4 = 0; h4 < 16; ++h4) {       // H=64 as 16 x float4
                    float4 kv = *(const float4*)(kr + h4 * 4);   // ds_load_b128
                    float4 qv = *(const float4*)(qr + h4 * 4);   // LDS broadcast
                    float4 w4 = *(const float4*)&swv[h4 * 4];    // LDS broadcast
                    s += fast_tanh(qv.x + kv.x) * w4.x;
                    s += fast_tanh(qv.y + kv.y) * w4.y;
                    s += fast_tanh(qv.z + kv.z) * w4.z;
                    s += fast_tanh(qv.w + kv.w) * w4.w;
                }
                sc[rr][c * 4 + i] = s;
            }
        }
    }

    int len = valid_lens[b];
    for (int rr = 0; rr < 2; ++rr) {
        float mx = MASK_VAL;
        #pragma unroll
        for (int j = 0; j < 16; ++j) {
            int k = ((j >> 2) * 128) + ((j & 3) * 32) + lane;
            if (k >= len) sc[rr][j] = MASK_VAL;
            mx = fmaxf(mx, sc[rr][j]);
        }
        #pragma unroll
        for (int off = 16; off >= 1; off >>= 1)
            mx = fmaxf(mx, __shfl_xor(mx, off, 32));

        float sum = 0.0f;
        #pragma unroll
        for (int j = 0; j < 16; ++j) {
            float e = __expf(sc[rr][j] - mx);
            sc[rr][j] = e;
            sum += e;
        }
        #pragma unroll
        for (int off = 16; off >= 1; off >>= 1)
            sum += __shfl_xor(sum, off, 32);
        float inv = 1.0f / sum;

        float* arow = attn + ((size_t)(b * Q_) + qt * 16 + (row0 + rr)) * K_;
        #pragma unroll
        for (int j = 0; j < 16; ++j) {
            int k = ((j >> 2) * 128) + ((j & 3) * 32) + lane;
            arow[k] = sc[rr][j] * inv;      // coalesced: lanes -> consecutive k
        }
    }
}

// ---------------------------------------------------------------------------
// Kernel 3: out = attn @ values via V_WMMA_F32_16X16X4_F32, fully LDS-fed.
// Block = 4 waves = 16(q) x 64(v) output tile. attn tile (16x512) staged once
// per block; values staged per 64-deep K chunk. All staging via async DMA.
// ---------------------------------------------------------------------------
__global__ __launch_bounds__(128)
void av_kernel(const float* __restrict__ attn,
               const float* __restrict__ values,
               float* __restrict__ out)
{
    __shared__ float sa[16 * APAD];     // attn tile   [16][512] padded (33 KB)
    __shared__ float sv[64 * VPAD];     // values tile [64][64]  padded (17 KB)

    int b  = blockIdx.z;
    int qt = blockIdx.y;
    int ng = blockIdx.x;
    int tid  = threadIdx.x;
    int lane = tid & 31;
    int wave = tid >> 5;
    int mn    = lane & 15;
    int khalf = (lane >> 4) << 1;
    int nb    = wave * 16;

    const float* Ag = attn   + ((size_t)(b * Q_) + qt * 16) * K_;
    const float* Vg = values + (size_t)b * K_ * D_ + ng * 64;

    // Stage attn tile: 16 rows x 128 float4 = 2048 copies / 128 thr = 16 each.
    for (int idx = tid; idx < 16 * 128; idx += 128) {
        int r = idx >> 7, j = idx & 127;
        stage16(Ag + (size_t)r * K_ + j * 4, &sa[r * APAD + j * 4]);
    }

    v8f acc = {};
    for (int c = 0; c < 8; ++c) {       // K = 512 in chunks of 64
        __syncthreads();                // protect previous chunk's sv reads
        #pragma unroll
        for (int idx = tid; idx < 64 * 16; idx += 128) {   // 8 copies/thread
            int r = idx >> 4, j = idx & 15;
            stage16(Vg + (size_t)(c * 64 + r) * D_ + j * 4,
                    &sv[r * VPAD + j * 4]);
        }
        stage_commit();                 // s_wait_asynccnt 0 + barrier

        #pragma unroll
        for (int kk = 0; kk < 64; kk += 4) {
            int kg = c * 64 + kk;
            v2f a = *(const v2f*)&sa[mn * APAD + kg + khalf];    // ds_load_b64
            v2f bv;
            bv.x = sv[(kk + khalf)     * VPAD + nb + mn];
            bv.y = sv[(kk + khalf + 1) * VPAD + nb + mn];
            acc = __builtin_amdgcn_wmma_f32_16x16x4_f32(
                      false, a, false, bv, (short)0, acc, false, false);
        }
    }

    float* O = out + ((size_t)(b * Q_) + qt * 16) * D_ + ng * 64 + nb + mn;
    #pragma unroll
    for (int r = 0; r < 8; ++r) {
        int m = r + ((lane >> 4) << 3);
        O[(size_t)m * D_] = acc[r];
    }
}

// ---------------------------------------------------------------------------
extern "C" void kernel_launch(void* const* d_in, const int* in_sizes, int n_in,
                              void* d_out, int out_size, void* d_ws, size_t ws_size,
                              hipStream_t stream)
{
    (void)in_sizes; (void)n_in; (void)out_size; (void)ws_size;

    const float* queries = (const float*)d_in[0];   // [B,Q,256]
    const float* keys    = (const float*)d_in[1];   // [B,K,256]
    const float* values  = (const float*)d_in[2];   // [B,K,256]
    const int*   vlens   = (const int*)  d_in[3];   // [B]
    const float* Wq      = (const float*)d_in[4];   // [64,256]
    const float* Wk      = (const float*)d_in[5];   // [64,256]
    const float* wv      = (const float*)d_in[6];   // [64]
    float* out = (float*)d_out;                     // [B,Q,256]

    // workspace: qf (2MB) | kf (2MB) | attn (16MB)
    char* ws = (char*)d_ws;
    float* qf   = (float*)ws;
    float* kf   = (float*)(ws + (size_t)B_ * Q_ * H_ * sizeof(float));
    float* attn = (float*)(ws + (size_t)B_ * Q_ * H_ * sizeof(float)
                              + (size_t)B_ * K_ * H_ * sizeof(float));

    proj_kernel<<<dim3((B_ * Q_) / 16 + (B_ * K_) / 16), 128, 0, stream>>>(
        queries, keys, Wq, Wk, qf, kf);

    score_softmax_kernel<<<dim3(B_ * (Q_ / 16)), 256, 0, stream>>>(
        qf, kf, wv, vlens, attn);

    av_kernel<<<dim3(D_ / 64, Q_ / 16, B_), 128, 0, stream>>>(
        attn, values, out);
}